// SRWMLayer_20615843021145
// MI455X (gfx1250) — compile-verified
//
#include <hip/hip_runtime.h>
#include <hip/hip_bf16.h>

#define L_SEQ 2048
#define BATCH 8
#define NH    16
#define EDIM  64
#define DDIM  1024
#define LN_EPS 1e-5f

typedef __attribute__((ext_vector_type(2))) float v2f;
typedef __attribute__((ext_vector_type(8))) float v8f;

// ---------------------------------------------------------------------------
// Kernel 1: SRWM sequential scan. One workgroup per (b,h) chain (128 WGs).
// 256 threads: thread (m = tid&63, c = tid>>6) owns wy/wq/wk[e, m] for
// e in [16c, 16c+16) in registers (48 state VGPRs). Row-sums of wq/wk are
// maintained incrementally (rank-1 update => rowsum[e] += k[e]*sum(delta)).
// ---------------------------------------------------------------------------
__global__ __launch_bounds__(256) void srwm_scan_kernel(
    const float* __restrict__ hidden,
    const float* __restrict__ wy0, const float* __restrict__ wq0,
    const float* __restrict__ wk0, const float* __restrict__ wb0,
    float* __restrict__ ys)
{
    const int bh = blockIdx.x;
    const int b  = bh / NH;
    const int h  = bh % NH;
    const int tid   = threadIdx.x;
    const int m     = tid & 63;
    const int c     = tid >> 6;       // e-chunk 0..3
    const int ebase = c << 4;

    __shared__ float xtb[2][EDIM];        // ping-pong input vector
    __shared__ float qvec[EDIM], kvec[EDIM], qkv[EDIM];
    __shared__ float sq[EDIM], sk[EDIM];  // row sums of wq, wk
    __shared__ float wbl[EDIM * 4];       // wb[e][f]
    __shared__ float bpart[EDIM * 8];     // [e][sel*4+f] products
    __shared__ float bred[8 * 8];         // [sf][g] level-1 partials
    __shared__ float part[4][4][EDIM];    // [quantity][chunk][m]
    __shared__ float dvec[3][EDIM];       // raw y_diff, q_diff, k_diff
    __shared__ float scal[8];             // beta0..3, bdiff0..3
    __shared__ float wsq[2], wsk[2];      // per-wave sums of q_diff/k_diff

    const float* Wy = wy0 + h * EDIM * EDIM;
    const float* Wq = wq0 + h * EDIM * EDIM;
    const float* Wk = wk0 + h * EDIM * EDIM;
    const float* Wb = wb0 + h * EDIM * 4;

    float wy_r[16], wq_r[16], wk_r[16];
    #pragma unroll
    for (int i = 0; i < 16; ++i) {
        const int e = ebase + i;
        wy_r[i] = Wy[e * EDIM + m];
        wq_r[i] = Wq[e * EDIM + m];
        wk_r[i] = Wk[e * EDIM + m];
    }
    if (tid < EDIM) {
        float s1 = 0.f, s2 = 0.f;
        for (int mm = 0; mm < EDIM; ++mm) {
            s1 += Wq[tid * EDIM + mm];
            s2 += Wk[tid * EDIM + mm];
        }
        sq[tid] = s1; sk[tid] = s2;
        #pragma unroll
        for (int f = 0; f < 4; ++f) wbl[tid * 4 + f] = Wb[tid * 4 + f];
        xtb[0][tid] = hidden[(0 * BATCH + b) * DDIM + h * EDIM + tid];
    }
    __syncthreads();

    for (int l = 0; l < L_SEQ; ++l) {
        const float* xt = xtb[l & 1];

        // P1: softmax(xt * rowsum) for q (wave 0) and k (wave 1)
        if (tid < 64) {
            const int  lane = tid & 31;
            const bool isQ  = (tid < 32);
            const float* srow = isQ ? sq : sk;
            float a0 = xt[lane]      * srow[lane];
            float a1 = xt[lane + 32] * srow[lane + 32];
            float mx = fmaxf(a0, a1);
            #pragma unroll
            for (int off = 16; off >= 1; off >>= 1)
                mx = fmaxf(mx, __shfl_xor(mx, off, 32));
            float e0 = __expf(a0 - mx), e1 = __expf(a1 - mx);
            float s = e0 + e1;
            #pragma unroll
            for (int off = 16; off >= 1; off >>= 1)
                s += __shfl_xor(s, off, 32);
            const float inv = 1.0f / s;
            float* dst = isQ ? qvec : kvec;
            dst[lane]      = e0 * inv;
            dst[lane + 32] = e1 * inv;
        }
        __syncthreads();

        // P2: qk = q - k; stage beta / b_diff products
        if (tid < 64) {
            const float qk = qvec[tid] - kvec[tid];
            qkv[tid] = qk;
            const float xe = xt[tid];
            #pragma unroll
            for (int f = 0; f < 4; ++f) {
                const float w = wbl[tid * 4 + f];
                bpart[tid * 8 + f]     = xe * w;
                bpart[tid * 8 + 4 + f] = qk * w;
            }
        }
        __syncthreads();

        // P3: contraction partials over this thread's e-chunk + bred level-1
        {
            float py = 0.f, pyd = 0.f, pqd = 0.f, pkd = 0.f;
            #pragma unroll
            for (int i = 0; i < 16; ++i) {
                const int e = ebase + i;
                const float xe = xt[e], qk = qkv[e];
                py  += xe * wy_r[i];
                pyd += qk * wy_r[i];
                pqd += qk * wq_r[i];
                pkd += qk * wk_r[i];
            }
            part[0][c][m] = py;
            part[1][c][m] = pyd;
            part[2][c][m] = pqd;
            part[3][c][m] = pkd;
            if (tid < 64) {
                const int sf = tid & 7, g = tid >> 3;
                float acc = 0.f;
                #pragma unroll
                for (int i = 0; i < 8; ++i) acc += bpart[(g * 8 + i) * 8 + sf];
                bred[sf * 8 + g] = acc;
            }
        }
        __syncthreads();

        // P4: reduce partials; emit y; wave sums of q_diff/k_diff; beta/b_diff
        {
            const int qi = c;
            const float v = part[qi][0][m] + part[qi][1][m] +
                            part[qi][2][m] + part[qi][3][m];
            if (qi == 0) {
                ys[(l * BATCH + b) * DDIM + h * EDIM + m] = v;
            } else {
                dvec[qi - 1][m] = v;
                if (qi >= 2) {  // waves 4..7 are uniform over this branch
                    float s = v;
                    #pragma unroll
                    for (int off = 16; off >= 1; off >>= 1)
                        s += __shfl_xor(s, off, 32);
                    if ((tid & 31) == 0) {
                        const int half = m >> 5;
                        if (qi == 2) wsq[half] = s; else wsk[half] = s;
                    }
                }
            }
            if (tid < 8) {
                float acc = 0.f;
                #pragma unroll
                for (int g = 0; g < 8; ++g) acc += bred[tid * 8 + g];
                scal[tid] = (tid < 4) ? (1.0f / (1.0f + __expf(-acc))) : acc;
            }
            // prefetch next timestep's x into the other buffer
            if (tid >= 64 && tid < 128 && (l + 1) < L_SEQ) {
                const int e = tid - 64;
                xtb[(l + 1) & 1][e] =
                    hidden[((l + 1) * BATCH + b) * DDIM + h * EDIM + e];
            }
        }
        __syncthreads();

        // P5: rank-1 updates to register state + row sums + wb
        {
            const float b0 = scal[0], b1 = scal[1], b2 = scal[2], b3 = scal[3];
            const float dy = b0 * dvec[0][m];
            const float dq = b1 * dvec[1][m];
            const float dk = b2 * dvec[2][m];
            #pragma unroll
            for (int i = 0; i < 16; ++i) {
                const float kk = kvec[ebase + i];
                wy_r[i] += kk * dy;
                wq_r[i] += kk * dq;
                wk_r[i] += kk * dk;
            }
            if (tid < 64) {
                const float kk  = kvec[tid];
                const float Sdq = b1 * (wsq[0] + wsq[1]);
                const float Sdk = b2 * (wsk[0] + wsk[1]);
                sq[tid] += kk * Sdq;
                sk[tid] += kk * Sdk;
                #pragma unroll
                for (int f = 0; f < 4; ++f)
                    wbl[tid * 4 + f] += kk * (b3 * scal[4 + f]);
            }
        }
        __syncthreads();
    }
}

// ---------------------------------------------------------------------------
// Kernel 2: out = LayerNorm(hidden + ys @ w_out^T) using V_WMMA_F32_16X16X4_F32.
// Block = 16 output rows x full 1024 cols (so LN fuses in-block).
// 8 waves; wave w accumulates 8 16x16 N-tiles (cols [w*128, w*128+128)).
// A chunks staged in LDS; B fragments stream from L2-resident w_out.
// ---------------------------------------------------------------------------
__global__ __launch_bounds__(256) void proj_ln_kernel(
    const float* __restrict__ ysrc,      // [L*B, D]
    const float* __restrict__ hidden,    // [L*B, D]
    const float* __restrict__ wout,      // [D, D] (out = ys @ wout^T)
    const float* __restrict__ gamma,
    const float* __restrict__ lbeta,
    float* __restrict__ out)
{
    const int r0   = blockIdx.x * 16;
    const int tid  = threadIdx.x;
    const int lane = tid & 31;
    const int w    = tid >> 5;

    __shared__ float As[16][68];          // A chunk, padded stride
    __shared__ float proj[16][1028];      // GEMM result tile
    __shared__ float rsum[16][16], rsum2[16][16];
    __shared__ float rmean[16], rrstd[16];

    v8f acc[8] = {};

    const int arow = lane & 15;
    const int koff = (lane >> 4) << 1;    // 0 (lanes 0-15) or 2 (lanes 16-31)
    const int ncol = lane & 15;
    const int rbase = (lane >> 4) << 3;   // C/D rows: v or v+8

    for (int kc = 0; kc < DDIM; kc += 64) {
        // stage A chunk: 16 rows x 64 k
        {
            const int row = tid >> 4;
            const int k4  = (tid & 15) << 2;
            const float4 av =
                *(const float4*)(ysrc + (r0 + row) * DDIM + kc + k4);
            As[row][k4 + 0] = av.x; As[row][k4 + 1] = av.y;
            As[row][k4 + 2] = av.z; As[row][k4 + 3] = av.w;
        }
        __syncthreads();

        #pragma unroll 4
        for (int kk = 0; kk < 64; kk += 4) {
            const float2 av = *(const float2*)&As[arow][kk + koff];
            v2f a; a.x = av.x; a.y = av.y;
            #pragma unroll
            for (int j = 0; j < 8; ++j) {
                const int n0 = (w * 8 + j) * 16;
                const float2 bv = *(const float2*)(
                    wout + (n0 + ncol) * DDIM + kc + kk + koff);
                v2f bb; bb.x = bv.x; bb.y = bv.y;
                acc[j] = __builtin_amdgcn_wmma_f32_16x16x4_f32(
                    false, a, false, bb, (short)0, acc[j], false, false);
            }
        }
        __syncthreads();
    }

    // write C fragments to LDS: VGPR v -> row (v | v+8), lane&15 -> col
    #pragma unroll
    for (int j = 0; j < 8; ++j) {
        const int col = (w * 8 + j) * 16 + ncol;
        #pragma unroll
        for (int v = 0; v < 8; ++v)
            proj[rbase + v][col] = acc[j][v];
    }
    __syncthreads();

    // residual + per-row mean/var partials (16 threads per row, 64 cols each)
    {
        const int row = tid >> 4;
        const int seg = tid & 15;
        const float* hrow = hidden + (r0 + row) * DDIM;
        float s = 0.f, s2 = 0.f;
        #pragma unroll 4
        for (int u = 0; u < 64; u += 4) {
            const int col = seg * 64 + u;
            const float4 hv = *(const float4*)(hrow + col);
            float v0 = proj[row][col + 0] + hv.x;
            float v1 = proj[row][col + 1] + hv.y;
            float v2 = proj[row][col + 2] + hv.z;
            float v3 = proj[row][col + 3] + hv.w;
            proj[row][col + 0] = v0; proj[row][col + 1] = v1;
            proj[row][col + 2] = v2; proj[row][col + 3] = v3;
            s  += v0 + v1 + v2 + v3;
            s2 += v0 * v0 + v1 * v1 + v2 * v2 + v3 * v3;
        }
        rsum[row][seg] = s; rsum2[row][seg] = s2;
    }
    __syncthreads();

    if (tid < 16) {
        float s = 0.f, s2 = 0.f;
        #pragma unroll
        for (int i = 0; i < 16; ++i) { s += rsum[tid][i]; s2 += rsum2[tid][i]; }
        const float mean = s * (1.0f / DDIM);
        const float var  = s2 * (1.0f / DDIM) - mean * mean;
        rmean[tid] = mean;
        rrstd[tid] = rsqrtf(var + LN_EPS);
    }
    __syncthreads();

    {
        const int row = tid >> 4;
        const int seg = tid & 15;
        const float mean = rmean[row], rstd = rrstd[row];
        float* orow = out + (r0 + row) * DDIM;
        #pragma unroll 4
        for (int u = 0; u < 64; u += 4) {
            const int col = seg * 64 + u;
            const float4 g  = *(const float4*)(gamma + col);
            const float4 bb = *(const float4*)(lbeta + col);
            float4 o;
            o.x = (proj[row][col + 0] - mean) * rstd * g.x + bb.x;
            o.y = (proj[row][col + 1] - mean) * rstd * g.y + bb.y;
            o.z = (proj[row][col + 2] - mean) * rstd * g.z + bb.z;
            o.w = (proj[row][col + 3] - mean) * rstd * g.w + bb.w;
            *(float4*)(orow + col) = o;
        }
    }
}

// ---------------------------------------------------------------------------
extern "C" void kernel_launch(void* const* d_in, const int* in_sizes, int n_in,
                              void* d_out, int out_size, void* d_ws, size_t ws_size,
                              hipStream_t stream)
{
    const float* hidden = (const float*)d_in[0];
    const float* wy     = (const float*)d_in[1];
    const float* wq     = (const float*)d_in[2];
    const float* wk     = (const float*)d_in[3];
    const float* wb     = (const float*)d_in[4];
    const float* wout   = (const float*)d_in[5];
    const float* gamma  = (const float*)d_in[6];
    const float* lbeta  = (const float*)d_in[7];
    float* out = (float*)d_out;
    float* ys  = (float*)d_ws;   // L*B*D fp32 = 64 MB scratch

    srwm_scan_kernel<<<BATCH * NH, 256, 0, stream>>>(hidden, wy, wq, wk, wb, ys);
    proj_ln_kernel<<<(L_SEQ * BATCH) / 16, 256, 0, stream>>>(
        ys, hidden, wout, gamma, lbeta, out);
}